// RegularizedCFMNet_47321949667542
// MI455X (gfx1250) — compile-verified
//
#include <hip/hip_runtime.h>
#include <hip/hip_bf16.h>
#include <math.h>

// ---------------- problem constants ----------------
#define N_DIM   40000
#define C_DIM   256
#define K_DIM   128
#define LAMBDA  1e-3
#define SLICES  16
#define SLICE_LEN (N_DIM / SLICES)   // 2500
#define KSTEPS    (SLICE_LEN / 4)    // 625 wmma k-steps per slice
#define FLUSH     25                 // flush f32 accum -> f64 every 25 steps (100 elems)
// waves: 2 matrices * 16 slices * 64 tile-pairs = 2048
#define WAVE_PART 1024               // doubles of partial output per wave (2 tiles)

typedef float v2f __attribute__((ext_vector_type(2)));
typedef float v8f __attribute__((ext_vector_type(8)));

// ---------------- complex helpers (fp64) ----------------
__device__ __forceinline__ double2 cmul(double2 a, double2 b) {
    return make_double2(a.x * b.x - a.y * b.y, a.x * b.y + a.y * b.x);
}
__device__ __forceinline__ double2 csub(double2 a, double2 b) {
    return make_double2(a.x - b.x, a.y - b.y);
}
__device__ __forceinline__ double2 cinv(double2 a) {
    double r = 1.0 / (a.x * a.x + a.y * a.y);
    return make_double2(a.x * r, -a.y * r);
}

// =====================================================================
// Kernel 1: project.  A = spec_grad_x @ feat_x, B = spec_grad_y @ feat_y
// complex128 x f32 via split-f32 WMMA (16x16x4 f32).  Each wave owns a
// 16x32 output patch (two adjacent 16x16 C tiles) of one matrix for one
// K-slice: one fp64->2xf32 split of the A fragment feeds 8 WMMAs.
// TN=2 keeps the wave inside the 256-VGPR direct window (no MSB copies).
// wave id = mat*1024 + slice*64 + tilePair   (2048 waves)
// =====================================================================
__global__ __launch_bounds__(256) void project_kernel(
    const float* __restrict__ feat_x, const float* __restrict__ feat_y,
    const double2* __restrict__ sgx,  const double2* __restrict__ sgy,
    double* __restrict__ partials)
{
    const int wave = blockIdx.x * 8 + (threadIdx.x >> 5);
    const int lane = threadIdx.x & 31;
    const int tile = wave & 63;          // tile-pair id
    const int s    = (wave >> 6) & (SLICES - 1);
    const int mat  = wave >> 10;

    const double2* spec = mat ? sgy : sgx;
    const float*   feat = mat ? feat_y : feat_x;

    const int tm  = tile & 7;    // 16-row band of K (8 bands)
    const int tnp = tile >> 3;   // 32-col band of C (8 bands)

    // WMMA f32 16x16x4 lane layout:
    //  A (16x4, M x K): lanes 0-15 -> M rows, vgpr0 = K {0|2}, vgpr1 = K {1|3}
    //  B (4x16, K x N): lanes (mod 16) -> N cols, vgpr pair = K rows
    const int mrow  = lane & 15;
    const int khalf = lane >> 4;
    const int ncol  = lane & 15;

    const double2* arow  = spec + (size_t)(tm * 16 + mrow) * N_DIM + 2 * khalf;
    const float*   fcol0 = feat + (size_t)(2 * khalf) * C_DIM + tnp * 32 + ncol;
    const float*   fcol1 = fcol0 + 16;

    const int n0 = s * SLICE_LEN;

    double d0re[8] = {0,0,0,0,0,0,0,0}, d0im[8] = {0,0,0,0,0,0,0,0};
    double d1re[8] = {0,0,0,0,0,0,0,0}, d1im[8] = {0,0,0,0,0,0,0,0};

    for (int blk = 0; blk < KSTEPS; blk += FLUSH) {
        v8f a0rh = {0,0,0,0,0,0,0,0}, a0rl = {0,0,0,0,0,0,0,0};
        v8f a0ih = {0,0,0,0,0,0,0,0}, a0il = {0,0,0,0,0,0,0,0};
        v8f a1rh = {0,0,0,0,0,0,0,0}, a1rl = {0,0,0,0,0,0,0,0};
        v8f a1ih = {0,0,0,0,0,0,0,0}, a1il = {0,0,0,0,0,0,0,0};
        #pragma unroll 2
        for (int t = 0; t < FLUSH; ++t) {
            const int n = n0 + (blk + t) * 4;
            // A fragment: two consecutive complex doubles (32 B / lane)
            double2 A0 = arow[n];
            double2 A1 = arow[n + 1];
            // Two B fragments (adjacent 16-col tiles), two f32 rows each
            float b00 = fcol0[(size_t)n * C_DIM];
            float b01 = fcol0[(size_t)(n + 1) * C_DIM];
            float b10 = fcol1[(size_t)n * C_DIM];
            float b11 = fcol1[(size_t)(n + 1) * C_DIM];
            v2f B0 = {b00, b01};
            v2f B1 = {b10, b11};
            // one 2-float split of the fp64 A fragment feeds 8 WMMAs
            float rh0 = (float)A0.x; float rl0 = (float)(A0.x - (double)rh0);
            float rh1 = (float)A1.x; float rl1 = (float)(A1.x - (double)rh1);
            float ih0 = (float)A0.y; float il0 = (float)(A0.y - (double)ih0);
            float ih1 = (float)A1.y; float il1 = (float)(A1.y - (double)ih1);
            v2f Arh = {rh0, rh1}, Arl = {rl0, rl1};
            v2f Aih = {ih0, ih1}, Ail = {il0, il1};
            // same-A pairs adjacent (operand reuse friendly)
            a0rh = __builtin_amdgcn_wmma_f32_16x16x4_f32(false, Arh, false, B0, (short)0, a0rh, false, false);
            a1rh = __builtin_amdgcn_wmma_f32_16x16x4_f32(false, Arh, false, B1, (short)0, a1rh, false, false);
            a0rl = __builtin_amdgcn_wmma_f32_16x16x4_f32(false, Arl, false, B0, (short)0, a0rl, false, false);
            a1rl = __builtin_amdgcn_wmma_f32_16x16x4_f32(false, Arl, false, B1, (short)0, a1rl, false, false);
            a0ih = __builtin_amdgcn_wmma_f32_16x16x4_f32(false, Aih, false, B0, (short)0, a0ih, false, false);
            a1ih = __builtin_amdgcn_wmma_f32_16x16x4_f32(false, Aih, false, B1, (short)0, a1ih, false, false);
            a0il = __builtin_amdgcn_wmma_f32_16x16x4_f32(false, Ail, false, B0, (short)0, a0il, false, false);
            a1il = __builtin_amdgcn_wmma_f32_16x16x4_f32(false, Ail, false, B1, (short)0, a1il, false, false);
        }
        #pragma unroll
        for (int e = 0; e < 8; ++e) {
            d0re[e] += (double)a0rh[e] + (double)a0rl[e];
            d0im[e] += (double)a0ih[e] + (double)a0il[e];
            d1re[e] += (double)a1rh[e] + (double)a1rl[e];
            d1im[e] += (double)a1ih[e] + (double)a1il[e];
        }
    }

    // per-wave partial patch: 2 subtiles x 32 lanes x 8 elems x (re,im)
    size_t base = (size_t)wave * WAVE_PART + (size_t)lane * 16;
    #pragma unroll
    for (int e = 0; e < 8; ++e) {
        partials[base + 2 * e]           = d0re[e];
        partials[base + 2 * e + 1]       = d0im[e];
        partials[base + 512 + 2 * e]     = d1re[e];
        partials[base + 512 + 2 * e + 1] = d1im[e];
    }
}

// =====================================================================
// Kernel 2: reduce slice partials -> Ahat/Bhat  (double2 [128][256])
// one thread per (mat, tilePair, subtile, lane, e) -> 65536 threads
// =====================================================================
__global__ __launch_bounds__(256) void reduce_kernel(
    const double* __restrict__ partials,
    double* __restrict__ Ahat, double* __restrict__ Bhat)
{
    int t = blockIdx.x * 256 + threadIdx.x;
    int e    = t & 7;
    int lane = (t >> 3) & 31;
    int sub  = (t >> 8) & 1;
    int tile = (t >> 9) & 63;
    int mat  = t >> 15;
    double re = 0.0, im = 0.0;
    for (int s = 0; s < SLICES; ++s) {
        size_t b = (size_t)(mat * 1024 + s * 64 + tile) * WAVE_PART
                 + (size_t)sub * 512 + (size_t)lane * 16;
        re += partials[b + 2 * e];
        im += partials[b + 2 * e + 1];
    }
    // C/D fragment layout: element (vgpr e, lane) = (M = e + 8*(lane/16), N = lane%16)
    int mloc = e + 8 * (lane >> 4);
    int nloc = lane & 15;
    int row = (tile & 7) * 16 + mloc;
    int col = ((tile >> 3) * 2 + sub) * 16 + nloc;
    double* dst = mat ? Bhat : Ahat;
    size_t o = ((size_t)row * C_DIM + col) * 2;
    dst[o]     = re;
    dst[o + 1] = im;
}

// =====================================================================
// Kernel 3: resolvent mask D[K][K]  (gamma = 0.5 -> sqrt)
// =====================================================================
__global__ __launch_bounds__(128) void mask_kernel(
    const double2* __restrict__ cex, const double2* __restrict__ cey,
    double* __restrict__ D)
{
    __shared__ double red[128];
    __shared__ double gxs[128], gys[128];
    __shared__ double sc;
    int i = threadIdx.x;
    double2 x = cex[i], y = cey[i];
    double ax = sqrt(x.x * x.x + x.y * x.y);
    double ay = sqrt(y.x * y.x + y.y * y.y);
    red[i] = fmax(ax, ay);
    __syncthreads();
    for (int off = 64; off >= 1; off >>= 1) {
        if (i < off) red[i] = fmax(red[i], red[i + off]);
        __syncthreads();
    }
    if (i == 0) sc = red[0];
    __syncthreads();
    double scale = sc;
    gxs[i] = sqrt(ax / scale);   // (|e|/scale)^0.5
    gys[i] = sqrt(ay / scale);
    __syncthreads();
    double gy = gys[i];
    double fy = gy / (gy * gy + 1.0);
    double hy = 1.0 / (gy * gy + 1.0);
    for (int j = 0; j < 128; ++j) {
        double gx = gxs[j];
        double fx = gx / (gx * gx + 1.0);
        double hx = 1.0 / (gx * gx + 1.0);
        double mr = fy - fx, mi = hy - hx;
        D[i * 128 + j] = mr * mr + mi * mi;
    }
}

// =====================================================================
// Kernel 4: gram.  AAt = A A^H, BAt = B A^H   (128x128 complex fp64)
// one thread per output element (i,j); C=256 reduction; L2-resident.
// =====================================================================
__global__ __launch_bounds__(256) void gram_kernel(
    const double* __restrict__ Ahat, const double* __restrict__ Bhat,
    double* __restrict__ AAt, double* __restrict__ BAt)
{
    int t = blockIdx.x * 256 + threadIdx.x;   // 16384 threads
    int i = t >> 7, j = t & 127;
    const double2* A = (const double2*)Ahat;
    const double2* B = (const double2*)Bhat;
    double ar = 0, ai = 0, br = 0, bi = 0;
    for (int c = 0; c < C_DIM; ++c) {
        double2 aj = A[(size_t)j * C_DIM + c];
        double2 av = A[(size_t)i * C_DIM + c];
        double2 bv = B[(size_t)i * C_DIM + c];
        ar = fma(av.x, aj.x, fma(av.y, aj.y, ar));
        ai = fma(av.y, aj.x, fma(-av.x, aj.y, ai));
        br = fma(bv.x, aj.x, fma(bv.y, aj.y, br));
        bi = fma(bv.y, aj.x, fma(-bv.x, aj.y, bi));
    }
    ((double2*)AAt)[t] = make_double2(ar, ai);
    ((double2*)BAt)[t] = make_double2(br, bi);
}

// =====================================================================
// Kernel 5: solve.  One workgroup per system i (128 WGs).
// M = AAt + lambda*diag(D[i,:]), solve M q = conj(BAt[i,:]),
// output Q[i,:] = conj(q).  Entire 128x129-padded complex128 matrix
// lives in LDS (264 KB) -- uses CDNA5's 320 KB WGP LDS.
// Partial-pivot LU + back substitution.
// =====================================================================
#define MSTRIDE 129
#define SOLVE_LDS_BYTES ((128 * MSTRIDE + 128) * (int)sizeof(double2))

__global__ __launch_bounds__(256) void solve_kernel(
    const double* __restrict__ AAt, const double* __restrict__ BAt,
    const double* __restrict__ D, double2* __restrict__ out)
{
    extern __shared__ double2 lds[];
    double2* M   = lds;                   // [128][129]
    double2* rhs = lds + 128 * MSTRIDE;   // [128]
    __shared__ double s_val[128];
    __shared__ int    s_idx[128];
    __shared__ double2 s_bcast;

    const int i   = blockIdx.x;   // system index
    const int tid = threadIdx.x;
    const double2* AA = (const double2*)AAt;
    const double2* BA = (const double2*)BAt;

    // build M and rhs
    for (int e = tid; e < 128 * 128; e += 256) {
        int r = e >> 7, c = e & 127;
        double2 v = AA[(size_t)r * 128 + c];
        if (r == c) v.x += LAMBDA * D[(size_t)i * 128 + r];
        M[r * MSTRIDE + c] = v;
    }
    for (int r = tid; r < 128; r += 256) {
        double2 b = BA[(size_t)i * 128 + r];
        rhs[r] = make_double2(b.x, -b.y);   // conj
    }
    __syncthreads();

    // ---------- forward elimination with partial pivoting ----------
    for (int k = 0; k < 127; ++k) {
        // pivot search over rows k..127
        if (tid < 128) {
            if (tid >= k) {
                double2 v = M[tid * MSTRIDE + k];
                s_val[tid] = v.x * v.x + v.y * v.y;
            } else {
                s_val[tid] = -1.0;
            }
            s_idx[tid] = tid;
        }
        __syncthreads();
        for (int off = 64; off >= 1; off >>= 1) {
            if (tid < off) {
                if (s_val[tid + off] > s_val[tid]) {
                    s_val[tid] = s_val[tid + off];
                    s_idx[tid] = s_idx[tid + off];
                }
            }
            __syncthreads();
        }
        int piv = s_idx[0];
        // row swap (active columns + rhs)
        if (piv != k) {
            for (int c = k + tid; c < 128; c += 256) {
                double2 tk = M[k * MSTRIDE + c];
                M[k * MSTRIDE + c]   = M[piv * MSTRIDE + c];
                M[piv * MSTRIDE + c] = tk;
            }
            if (tid == 0) {
                double2 tr = rhs[k]; rhs[k] = rhs[piv]; rhs[piv] = tr;
            }
        }
        __syncthreads();
        if (tid == 0) s_bcast = cinv(M[k * MSTRIDE + k]);
        __syncthreads();
        double2 pinv = s_bcast;
        double2 rk   = rhs[k];
        // multipliers + rhs update
        for (int r = k + 1 + tid; r < 128; r += 256) {
            double2 l = cmul(M[r * MSTRIDE + k], pinv);
            M[r * MSTRIDE + k] = l;
            rhs[r] = csub(rhs[r], cmul(l, rk));
        }
        __syncthreads();
        // trailing submatrix rank-1 update
        int nrc = 127 - k;
        int total = nrc * nrc;
        for (int idx = tid; idx < total; idx += 256) {
            int r = k + 1 + idx / nrc;
            int c = k + 1 + idx % nrc;
            M[r * MSTRIDE + c] =
                csub(M[r * MSTRIDE + c],
                     cmul(M[r * MSTRIDE + k], M[k * MSTRIDE + c]));
        }
        __syncthreads();
    }

    // ---------- back substitution (solution accumulates in rhs) ----------
    for (int k = 127; k >= 0; --k) {
        if (tid == 0) {
            double2 xk = cmul(rhs[k], cinv(M[k * MSTRIDE + k]));
            rhs[k] = xk;
            s_bcast = xk;
        }
        __syncthreads();
        double2 xk = s_bcast;
        for (int r = tid; r < k; r += 256) {
            rhs[r] = csub(rhs[r], cmul(M[r * MSTRIDE + k], xk));
        }
        __syncthreads();
    }

    // Q[i, j] = conj(x[j])
    for (int j = tid; j < 128; j += 256) {
        double2 x = rhs[j];
        out[(size_t)i * 128 + j] = make_double2(x.x, -x.y);
    }
}

// =====================================================================
// launcher
// =====================================================================
extern "C" void kernel_launch(void* const* d_in, const int* in_sizes, int n_in,
                              void* d_out, int out_size, void* d_ws, size_t ws_size,
                              hipStream_t stream) {
    const float*   feat_x = (const float*)d_in[0];
    const float*   feat_y = (const float*)d_in[1];
    const double2* sgx    = (const double2*)d_in[2];
    const double2* sgy    = (const double2*)d_in[3];
    const double2* cex    = (const double2*)d_in[4];
    const double2* cey    = (const double2*)d_in[5];

    double* ws = (double*)d_ws;
    // workspace layout (doubles):
    double* P    = ws;                                   // 2048 * 1024 = 2,097,152
    double* Ahat = P + (size_t)2048 * WAVE_PART;         // 65,536
    double* Bhat = Ahat + (size_t)2 * K_DIM * C_DIM;     // 65,536
    double* Dm   = Bhat + (size_t)2 * K_DIM * C_DIM;     // 16,384
    double* AAt  = Dm + (size_t)K_DIM * K_DIM;           // 32,768
    double* BAt  = AAt + (size_t)2 * K_DIM * K_DIM;      // 32,768

    // 1) spectral projection via split-f32 WMMA: 2048 waves, 8 waves/block
    project_kernel<<<256, 256, 0, stream>>>(feat_x, feat_y, sgx, sgy, P);
    // 2) slice reduction -> Ahat/Bhat
    reduce_kernel<<<256, 256, 0, stream>>>(P, Ahat, Bhat);
    // 3) resolvent mask
    mask_kernel<<<1, 128, 0, stream>>>(cex, cey, Dm);
    // 4) gram matrices
    gram_kernel<<<64, 256, 0, stream>>>(Ahat, Bhat, AAt, BAt);
    // 5) 128 pivoted-LU solves, whole matrix in 320KB WGP LDS
    solve_kernel<<<128, 256, SOLVE_LDS_BYTES, stream>>>(AAt, BAt, Dm, (double2*)d_out);
}